// FocusingLoss_47734266528022
// MI455X (gfx1250) — compile-verified
//
#include <hip/hip_runtime.h>

typedef float v2f __attribute__((ext_vector_type(2)));
typedef float v8f __attribute__((ext_vector_type(8)));

#define HH 512
#define WW 512
#define NQ 9   // 0:mse 1:tsum 2:t*y 3:t*x 4:psum 5:p*y 6:p*y2 7:p*x 8:p*x2

// ---------------------------------------------------------------- init
__global__ __launch_bounds__(256) void fl_init(float* ws, float* out, int nimg) {
    int i = blockIdx.x * 256 + threadIdx.x;
    if (i < nimg * NQ) ws[i] = 0.f;
    if (i == 0) out[0] = 0.f;
}

// ---------------------------------------------------------------- pass 1: moments via FP32 WMMA
// grid = nimg * 32 blocks; block = 256 threads = 8 waves.
// Each block: one 16-row band of one image. Each wave: 64 columns of that band.
// B weight matrix columns {1, x, x^2} maintained by forward differences (exact in f32).
__global__ __launch_bounds__(256) void fl_moments(const float* __restrict__ pred,
                                                  const float* __restrict__ targ,
                                                  float* __restrict__ ws) {
    const int img     = blockIdx.x >> 5;
    const int band    = blockIdx.x & 31;
    const int rowbase = band * 16;
    const int lane    = threadIdx.x & 31;
    const int wv      = threadIdx.x >> 5;
    const int m       = lane & 15;        // M index (A rows) == N index (B,D cols)
    const int khalf   = lane >> 4;        // selects K pair {0,1} vs {2,3}

    const float* pbase = pred + (size_t)img * (HH * WW);
    const float* tbase = targ + (size_t)img * (HH * WW);
    const size_t rowoff = (size_t)(rowbase + m) * WW;
    const float* prow = pbase + rowoff;
    const float* trow = tbase + rowoff;

    // weight-column selectors (N = m): col0 = 1, col1 = x, col2 = x^2
    const float f0 = (m == 0) ? 1.f : 0.f;
    const float f1 = (m == 1) ? 1.f : 0.f;
    const float f2 = (m == 2) ? 1.f : 0.f;

    // B fragment init at x = colbase + 2*khalf (+1 for .y), stride 4 per tile
    const int   colbase = wv * 64;
    const float xs  = (float)(colbase + 2 * khalf);
    const float xs1 = xs + 1.f;
    v2f b, db;
    b.x  = f0 + f1 * xs  + f2 * xs  * xs;
    b.y  = f0 + f1 * xs1 + f2 * xs1 * xs1;
    db.x = f1 * 4.f + f2 * (8.f * xs  + 16.f);     // w(x+4)-w(x)
    db.y = f1 * 4.f + f2 * (8.f * xs1 + 16.f);
    const float ddb = f2 * 32.f;                   // second difference of x^2

    v8f dp = {0.f, 0.f, 0.f, 0.f, 0.f, 0.f, 0.f, 0.f};
    v8f dt = {0.f, 0.f, 0.f, 0.f, 0.f, 0.f, 0.f, 0.f};
    v2f msev = {0.f, 0.f};

    const int kcol = colbase + 2 * khalf;
#pragma unroll 8
    for (int t = 0; t < 16; ++t) {
        const int xk = kcol + t * 4;

        // A fragments: lane holds P[row][xk], P[row][xk+1]  (16x4 f32 layout)
        v2f ap = *(const v2f*)(prow + xk);
        v2f at = *(const v2f*)(trow + xk);

        // one cacheline-ahead prefetch per 128B consumed (statically resolved)
        if ((t & 7) == 0) {
            __builtin_prefetch(prow + xk + 32, 0, 3);
            __builtin_prefetch(trow + xk + 32, 0, 3);
        }

        dp = __builtin_amdgcn_wmma_f32_16x16x4_f32(false, ap, false, b, (short)0, dp, false, false);
        dt = __builtin_amdgcn_wmma_f32_16x16x4_f32(false, at, false, b, (short)0, dt, false, false);

        // MSE: packed diff + packed FMA accumulate
        v2f d;
        d.x = ap.x - at.x;
        d.y = ap.y - at.y;
        msev.x += d.x * d.x;
        msev.y += d.y * d.y;

        // forward-difference B update (exact integer arithmetic in f32)
        b.x += db.x;  b.y += db.y;
        db.x += ddb;  db.y += ddb;
    }
    float mse = msev.x + msev.y;

    // Fold D rows with y-weights. Lane L owns M = r + 8*(L>>4), N = L&15.
    const float ybase = (float)(rowbase + 8 * khalf);
    float cs = 0.f, cys = 0.f, cyys = 0.f, ct = 0.f, cty = 0.f;
#pragma unroll
    for (int r = 0; r < 8; ++r) {
        const float yr = ybase + (float)r;
        cs   += dp[r];
        cys  += yr * dp[r];
        cyys += yr * yr * dp[r];
        ct   += dt[r];
        cty  += yr * dt[r];
    }

    // wave32 reductions / gathers
    for (int o = 16; o > 0; o >>= 1) mse += __shfl_down(mse, o, 32);
    const float q1 = __shfl(ct,   0, 32) + __shfl(ct,   16, 32); // sum t
    const float q2 = __shfl(cty,  0, 32) + __shfl(cty,  16, 32); // sum t*y
    const float q3 = __shfl(ct,   1, 32) + __shfl(ct,   17, 32); // sum t*x
    const float q4 = __shfl(cs,   0, 32) + __shfl(cs,   16, 32); // sum p
    const float q5 = __shfl(cys,  0, 32) + __shfl(cys,  16, 32); // sum p*y
    const float q6 = __shfl(cyys, 0, 32) + __shfl(cyys, 16, 32); // sum p*y^2
    const float q7 = __shfl(cs,   1, 32) + __shfl(cs,   17, 32); // sum p*x
    const float q8 = __shfl(cs,   2, 32) + __shfl(cs,   18, 32); // sum p*x^2

    __shared__ float smem[8][NQ];
    if (lane == 0) {
        smem[wv][0] = mse; smem[wv][1] = q1; smem[wv][2] = q2;
        smem[wv][3] = q3;  smem[wv][4] = q4; smem[wv][5] = q5;
        smem[wv][6] = q6;  smem[wv][7] = q7; smem[wv][8] = q8;
    }
    __syncthreads();
    if (threadIdx.x < NQ) {
        float a = 0.f;
#pragma unroll
        for (int w2 = 0; w2 < 8; ++w2) a += smem[w2][threadIdx.x];
        atomicAdd(&ws[img * NQ + threadIdx.x], a);
    }
}

// ---------------------------------------------------------------- pass 2: centroid window + assemble loss
__global__ __launch_bounds__(256) void fl_finalize(const float* __restrict__ pred,
                                                   const float* __restrict__ ws,
                                                   float* __restrict__ out, int nimg) {
    const int img = blockIdx.x;
    const float tsum = ws[img * NQ + 1];
    const float tys  = ws[img * NQ + 2];
    const float txs  = ws[img * NQ + 3];
    const float psum = ws[img * NQ + 4];
    const float pys  = ws[img * NQ + 5];
    const float pyys = ws[img * NQ + 6];
    const float pxs  = ws[img * NQ + 7];
    const float pxxs = ws[img * NQ + 8];

    const bool  valid = tsum >= 1e-8f;
    const float tss   = valid ? tsum : 1.f;
    const float cy    = tys / tss;
    const float cx    = txs / tss;
    const bool  ptok  = psum > 1e-8f;
    const float pts   = ptok ? psum : 1.f;
    const bool  active = valid && ptok;

    float fe = 0.f;
    if (active) {                                    // block-uniform branch
        const int y0 = max(0,      (int)ceilf(cy - 20.f));
        const int y1 = min(HH - 1, (int)floorf(cy + 20.f));
        const int x0 = max(0,      (int)ceilf(cx - 20.f));
        const int x1 = min(WW - 1, (int)floorf(cx + 20.f));
        const int wX = x1 - x0 + 1;
        const int nWin = (y1 - y0 + 1) * wX;
        const float* p = pred + (size_t)img * (HH * WW);
        for (int i = threadIdx.x; i < nWin; i += 256) {
            const int y = y0 + i / wX;
            const int x = x0 + i % wX;
            const float dy = (float)y - cy, dx = (float)x - cx;
            if (dy * dy + dx * dx <= 400.f) fe += p[(size_t)y * WW + x];
        }
    }
    for (int o = 16; o > 0; o >>= 1) fe += __shfl_down(fe, o, 32);
    __shared__ float sred[8];
    if ((threadIdx.x & 31) == 0) sred[threadIdx.x >> 5] = fe;
    __syncthreads();
    if (threadIdx.x == 0) {
        float fet = 0.f;
#pragma unroll
        for (int w = 0; w < 8; ++w) fet += sred[w];
        const float spread = (pyys - 2.f * cy * pys + cy * cy * psum)
                           + (pxxs - 2.f * cx * pxs + cx * cx * psum);
        const float r = 1.f - fet / pts;
        float contrib = active ? (10.f * r * r + spread / pts) : 0.f;
        if (img == 0) {  // base MSE term, added exactly once
            float mt = 0.f;
            for (int i = 0; i < nimg; ++i) mt += ws[i * NQ + 0];
            contrib += mt * (1.f / ((float)nimg * (float)(HH * WW)));
        }
        atomicAdd(out, contrib);
    }
}

// ---------------------------------------------------------------- launch
extern "C" void kernel_launch(void* const* d_in, const int* in_sizes, int n_in,
                              void* d_out, int out_size, void* d_ws, size_t ws_size,
                              hipStream_t stream) {
    const float* pred = (const float*)d_in[0];
    const float* targ = (const float*)d_in[1];
    float* out = (float*)d_out;
    float* ws  = (float*)d_ws;
    const int nimg = in_sizes[0] / (HH * WW);   // 128 for [16,8,512,512]

    fl_init<<<(nimg * NQ + 255) / 256, 256, 0, stream>>>(ws, out, nimg);
    fl_moments<<<nimg * 32, 256, 0, stream>>>(pred, targ, ws);
    fl_finalize<<<nimg, 256, 0, stream>>>(pred, ws, out, nimg);
}